// GCN_26989574488548
// MI455X (gfx1250) — compile-verified
//
#include <hip/hip_runtime.h>

// Problem constants (fixed by the reference).
#define N_NODES 50000
#define N_EDGES 800000
#define FDIM    128
#define HDIM    128
#define CDIM    2

// LDS row stride for the 16x128 A tile: 132 floats -> bank index
// (4*row + k) mod 64, conflict-free for the WMMA fragment read pattern.
#define LDS_STRIDE 132

typedef __attribute__((ext_vector_type(2))) float v2f;
typedef __attribute__((ext_vector_type(8))) float v8f;

__device__ __forceinline__ float atomic_add_f32(float* p, float v) {
  return __hip_atomic_fetch_add(p, v, __ATOMIC_RELAXED, __HIP_MEMORY_SCOPE_AGENT);
}

// ---------------------------------------------------------------- utilities
__global__ void zero_f32(float* __restrict__ p, int n) {
  int i = blockIdx.x * blockDim.x + threadIdx.x;
  if (i < n) p[i] = 0.0f;
}

// agg[i] = bias[i % 128]  (aggregation buffer pre-seeded with the bias)
__global__ void bias_init_kernel(float* __restrict__ out, const float* __restrict__ bias) {
  int i = blockIdx.x * blockDim.x + threadIdx.x;
  if (i < N_NODES * HDIM) out[i] = bias[i & (HDIM - 1)];
}

// deg[d] = sum of edge weights into d (+1.0 per node for the self-loop)
__global__ void deg_kernel(const int* __restrict__ dst, const float* __restrict__ ew,
                           float* __restrict__ deg) {
  int i = blockIdx.x * blockDim.x + threadIdx.x;
  if (i < N_EDGES) {
    atomic_add_f32(&deg[dst[i]], ew[i]);
  } else if (i < N_EDGES + N_NODES) {
    atomic_add_f32(&deg[i - N_EDGES], 1.0f);   // self-loop weight 1
  }
}

// in-place: deg -> D^{-1/2}
__global__ void dinv_kernel(float* __restrict__ deg) {
  int i = blockIdx.x * blockDim.x + threadIdx.x;
  if (i < N_NODES) {
    float d = deg[i];
    deg[i] = (d > 0.0f) ? rsqrtf(d) : 0.0f;
  }
}

// norm[e] = dinv[src]*ew*dinv[dst]  (self-loop norm = dinv^2, computed on the fly)
__global__ void norm_kernel(const int* __restrict__ src, const int* __restrict__ dst,
                            const float* __restrict__ ew, const float* __restrict__ dinv,
                            float* __restrict__ norm) {
  int e = blockIdx.x * blockDim.x + threadIdx.x;
  if (e < N_EDGES) norm[e] = dinv[src[e]] * ew[e] * dinv[dst[e]];
}

// ---------------------------------------------------------- WMMA fp32 GEMM
// Out[N,128] = op(A)[N,128] @ W[128,128]  (+bias/relu on store per `fuse`)
// fuse bit0: add bias on store   bit1: relu on store   bit2: relu on A load
//
// Block = 256 threads = 8 waves; block computes a 16-row x 128-col tile.
// The 16x128 A tile is staged once into LDS (coalesced float4), then each
// wave owns one 16x16 output tile fed by V_WMMA_F32_16X16X4_F32, 32 K-steps.
// Fragment layouts per CDNA5 ISA 7.12.2:
//   A 16x4 : M = lane&15, K = vgpr + 2*(lane>=16)   (ds_load_b64 from LDS)
//   B 4x16 : N = lane&15, K = vgpr + 2*(lane>=16)
//   C 16x16: N = lane&15, M = vgpr + 8*(lane>=16)
__global__ __launch_bounds__(256) void gemm128_wmma(const float* __restrict__ A,
                                                    const float* __restrict__ W,
                                                    const float* __restrict__ bias,
                                                    float* __restrict__ Out,
                                                    int fuse) {
  __shared__ float As[16 * LDS_STRIDE];

  const int t       = threadIdx.x;
  const int lane    = t & 31;
  const int wave    = t >> 5;               // 0..7 -> column tile
  const int laneLo  = lane & 15;
  const int laneHi  = lane >> 4;            // 0 or 1
  const int rowBase = blockIdx.x * 16;      // grid = N/16 exactly (50000/16=3125)
  const int colBase = wave * 16;

  // --- cooperative A-tile stage: 16 rows x 128 cols, 2x float4 per thread ---
  {
    const int r  = t >> 4;                  // 0..15
    const int c0 = (t & 15) * 4;            // 0..60 step 4
    const float* srcp = A + (long long)(rowBase + r) * HDIM;
    float4 v0 = *(const float4*)(srcp + c0);
    float4 v1 = *(const float4*)(srcp + c0 + 64);
    if (fuse & 4) {                         // fused ReLU on the A operand
      v0.x = fmaxf(v0.x, 0.f); v0.y = fmaxf(v0.y, 0.f);
      v0.z = fmaxf(v0.z, 0.f); v0.w = fmaxf(v0.w, 0.f);
      v1.x = fmaxf(v1.x, 0.f); v1.y = fmaxf(v1.y, 0.f);
      v1.z = fmaxf(v1.z, 0.f); v1.w = fmaxf(v1.w, 0.f);
    }
    float* dp = As + r * LDS_STRIDE;        // 528B row stride: 16B aligned
    *(float4*)(dp + c0)      = v0;
    *(float4*)(dp + c0 + 64) = v1;
  }
  __syncthreads();

  const float* arow = As + laneLo * LDS_STRIDE;
  const float* wcol = W + colBase + laneLo;

  v8f acc = {};
#pragma unroll
  for (int k = 0; k < HDIM; k += 4) {
    const int ka = k + laneHi * 2;
    v2f af, bf;
    af.x = arow[ka];                        // ds_load_b64 (8B aligned, no bank conflicts)
    af.y = arow[ka + 1];
    bf.x = wcol[(long long)ka * HDIM];
    bf.y = wcol[(long long)(ka + 1) * HDIM];
    acc = __builtin_amdgcn_wmma_f32_16x16x4_f32(false, af, false, bf,
                                                (short)0, acc, false, false);
  }

  const float bv = (fuse & 1) ? bias[colBase + laneLo] : 0.0f;
#pragma unroll
  for (int i = 0; i < 8; ++i) {
    const int m = i + 8 * laneHi;
    float v = acc[i] + bv;
    if (fuse & 2) v = fmaxf(v, 0.0f);
    Out[(long long)(rowBase + m) * HDIM + colBase + laneLo] = v;
  }
}

// -------------------------------------------------- edge gather/scatter-add
// One 32-lane group per (edge or self-loop): lane holds float4 of the 128-wide
// feature row.  out[dst,:] += xw[src,:] * norm   via hardware f32 atomics.
__global__ void scatter_kernel(const float* __restrict__ xw, const float* __restrict__ norm,
                               const int* __restrict__ src, const int* __restrict__ dst,
                               const float* __restrict__ dinv, float* __restrict__ out) {
  const unsigned gid  = blockIdx.x * blockDim.x + threadIdx.x;
  const unsigned e    = gid >> 5;
  const unsigned lane = gid & 31;
  if (e >= (unsigned)(N_EDGES + N_NODES)) return;

  int s, d;
  float nm;
  if (e < (unsigned)N_EDGES) {
    s = src[e]; d = dst[e]; nm = norm[e];
  } else {
    s = d = (int)(e - N_EDGES);
    const float di = dinv[s];
    nm = di * di;                             // self-loop: dinv * 1 * dinv
  }

  const float4 v = ((const float4*)(xw + (long long)s * HDIM))[lane];
  float* o = out + (long long)d * HDIM + lane * 4;
  atomic_add_f32(o + 0, v.x * nm);
  atomic_add_f32(o + 1, v.y * nm);
  atomic_add_f32(o + 2, v.z * nm);
  atomic_add_f32(o + 3, v.w * nm);
}

// final tiny GEMM, one wave per node:
//   out[n, 0..1] = relu(h[n,:]) @ W_out + b_out
// Lane k4 covers features 4*k4..4*k4+3 (coalesced float4), then a 5-step
// wave32 shuffle-xor reduction produces both outputs.
__global__ __launch_bounds__(256) void out_gemm_kernel(const float* __restrict__ h,
                                                       const float* __restrict__ Wout,
                                                       const float* __restrict__ bout,
                                                       float* __restrict__ out) {
  const int lane = threadIdx.x & 31;
  const int node = blockIdx.x * 8 + (threadIdx.x >> 5);   // launched with 196 blocks (nodes 0..1567)
  if (node >= N_NODES) return;

  float4 v = ((const float4*)(h + (long long)node * HDIM))[lane];
  v.x = fmaxf(v.x, 0.f); v.y = fmaxf(v.y, 0.f);
  v.z = fmaxf(v.z, 0.f); v.w = fmaxf(v.w, 0.f);

  const float* wr = Wout + lane * 4 * CDIM;               // [128,2] row-major
  float a0 = v.x * wr[0] + v.y * wr[2] + v.z * wr[4] + v.w * wr[6];
  float a1 = v.x * wr[1] + v.y * wr[3] + v.z * wr[5] + v.w * wr[7];

#pragma unroll
  for (int off = 16; off > 0; off >>= 1) {
    a0 += __shfl_xor(a0, off, 32);
    a1 += __shfl_xor(a1, off, 32);
  }
  if (lane == 0) {
    out[node * CDIM + 0] = a0 + bout[0];
    out[node * CDIM + 1] = a1 + bout[1];
  }
}

// --------------------------------------------------------------- dispatcher
extern "C" void kernel_launch(void* const* d_in, const int* in_sizes, int n_in,
                              void* d_out, int out_size, void* d_ws, size_t ws_size,
                              hipStream_t stream) {
  (void)in_sizes; (void)n_in; (void)out_size; (void)ws_size;

  const float* x     = (const float*)d_in[0];
  const int*   eidx  = (const int*)d_in[1];   // [2, E] flattened
  const float* ew    = (const float*)d_in[2];
  const float* W_in  = (const float*)d_in[3];
  const float* b_in  = (const float*)d_in[4];
  const float* W0    = (const float*)d_in[5];
  const float* b0    = (const float*)d_in[6];
  const float* W1    = (const float*)d_in[7];
  const float* b1    = (const float*)d_in[8];
  const float* W_out = (const float*)d_in[9];
  const float* b_out = (const float*)d_in[10];
  float* out = (float*)d_out;

  const int* src = eidx;
  const int* dst = eidx + N_EDGES;

  // Workspace layout (floats): dinv[N] | norm[E] | bufA[N*H] | bufB[N*H] | bufC[N*H]
  float* dinv = (float*)d_ws;
  float* norm = dinv + N_NODES;
  float* bufA = norm + N_EDGES;
  float* bufB = bufA + (size_t)N_NODES * HDIM;
  float* bufC = bufB + (size_t)N_NODES * HDIM;

  const int T = 256;
  const int nBlkN   = (N_NODES + T - 1) / T;
  const int nBlkE   = (N_EDGES + T - 1) / T;
  const int nBlkEN  = (N_EDGES + N_NODES + T - 1) / T;
  const int nBlkNH  = (N_NODES * HDIM + T - 1) / T;
  const int nBlkSct = ((N_EDGES + N_NODES) * 32 + T - 1) / T;
  const dim3 gGrid(N_NODES / 16);           // 3125 row-tiles, exact

  // --- normalization coefficients (shared by both conv layers) ---
  zero_f32<<<nBlkN, T, 0, stream>>>(dinv, N_NODES);
  deg_kernel<<<nBlkEN, T, 0, stream>>>(dst, ew, dinv);
  dinv_kernel<<<nBlkN, T, 0, stream>>>(dinv);
  norm_kernel<<<nBlkE, T, 0, stream>>>(src, dst, ew, dinv, norm);

  // --- input layer: bufA = relu(x @ W_in + b_in) ---
  gemm128_wmma<<<gGrid, 256, 0, stream>>>(x, W_in, b_in, bufA, /*bias|relu on store*/3);

  // --- conv layer 0: bufC = b0 + Scatter(bufA @ W0) ---
  gemm128_wmma<<<gGrid, 256, 0, stream>>>(bufA, W0, nullptr, bufB, 0);
  bias_init_kernel<<<nBlkNH, T, 0, stream>>>(bufC, b0);
  scatter_kernel<<<nBlkSct, T, 0, stream>>>(bufB, norm, src, dst, dinv, bufC);

  // --- conv layer 1: bufB = b1 + Scatter(relu(bufC) @ W1) ---
  gemm128_wmma<<<gGrid, 256, 0, stream>>>(bufC, W1, nullptr, bufA, /*relu on load*/4);
  bias_init_kernel<<<nBlkNH, T, 0, stream>>>(bufB, b1);
  scatter_kernel<<<nBlkSct, T, 0, stream>>>(bufA, norm, src, dst, dinv, bufB);

  // --- output layer: out = relu(bufB) @ W_out + b_out ---
  out_gemm_kernel<<<nBlkN, T, 0, stream>>>(bufB, W_out, b_out, out);
}